// LearnedAttentionPooling_29454885716445
// MI455X (gfx1250) — compile-verified
//
#include <hip/hip_runtime.h>
#include <math.h>

typedef __attribute__((ext_vector_type(2))) float v2f;
typedef __attribute__((ext_vector_type(8))) float v8f;

#define B_MAX      32
#define N_TOK      4096
#define IN_DIM     64
#define FEAT_DIM   768
#define NQ         4
#define NEG_BIG    (-3.402823466e38f)   // finfo(float32).min, matches jax masking

// ---------------------------------------------------------------------------
// Kernel 1: q_proj[q][i] = sum_f query[q][f] * W[f][i]   (Q=4, F=768, I=64)
//           q_bias[q]    = sum_f query[q][f] * b[f]
// One block of 256 threads; trivial cost.
// ---------------------------------------------------------------------------
__global__ void lap_qproj_kernel(const float* __restrict__ qt,   // [4][768]
                                 const float* __restrict__ W,    // [768][64]
                                 const float* __restrict__ bias, // [768]
                                 float* __restrict__ qp,         // [4][64]
                                 float* __restrict__ qb)         // [4]
{
    const int t = threadIdx.x;          // 0..255
    const int q = t >> 6;
    const int i = t & 63;
    float acc = 0.0f;
    const float* qrow = qt + q * FEAT_DIM;
    for (int f = 0; f < FEAT_DIM; ++f)
        acc += qrow[f] * W[f * IN_DIM + i];   // lanes read consecutive i -> coalesced
    qp[q * IN_DIM + i] = acc;
    if (t < NQ) {
        float s = 0.0f;
        const float* qr = qt + t * FEAT_DIM;
        for (int f = 0; f < FEAT_DIM; ++f) s += qr[f] * bias[f];
        qb[t] = s;
    }
}

// ---------------------------------------------------------------------------
// Kernel 2: one block per (b, q). Fused: scores -> softmax -> 64-dim pooling.
//   scores[n] = q_proj[q] . x[b,n] + q_bias[q]  (masked -> -FLT_MAX)
//   w = softmax(scores);  pooled[b,q,:] = sum_n w[n] * x[b,n,:]
// LDS: 16KB score/weight buffer + small reduction scratch. 256 threads.
// x[b] is 1 MB -> L2 resident between the two passes.
// ---------------------------------------------------------------------------
__global__ void lap_pool_kernel(const float* __restrict__ x,    // [B][4096][64]
                                const int*   __restrict__ mask, // [B][4096]
                                const float* __restrict__ qp,   // [4][64]
                                const float* __restrict__ qb,   // [4]
                                float* __restrict__ pooled)     // [B*4][64]
{
    const int b = blockIdx.x;
    const int q = blockIdx.y;
    const int t = threadIdx.x;          // 0..255

    __shared__ float s_w[N_TOK];        // scores, then unnormalized weights
    __shared__ float s_qp[IN_DIM];
    __shared__ float s_red[256];
    __shared__ float s_part[4][IN_DIM];

    if (t < IN_DIM) s_qp[t] = qp[q * IN_DIM + t];
    __syncthreads();

    const float qbias = qb[q];
    const float* xb = x + (size_t)b * N_TOK * IN_DIM;
    const int*   mb = mask + (size_t)b * N_TOK;

    // ---- Pass A: scores (coalesced float4 loads; 16 tokens per thread) ----
    float lmax = NEG_BIG;
    for (int n = t; n < N_TOK; n += 256) {
        const float* xr = xb + n * IN_DIM;   // 256B-aligned row
        float s = qbias;
        #pragma unroll
        for (int i = 0; i < IN_DIM; i += 4) {
            float4 xv = *reinterpret_cast<const float4*>(xr + i);
            s += s_qp[i]     * xv.x + s_qp[i + 1] * xv.y
               + s_qp[i + 2] * xv.z + s_qp[i + 3] * xv.w;
        }
        if (mb[n] == 0) s = NEG_BIG;
        s_w[n] = s;
        lmax = fmaxf(lmax, s);
    }

    // ---- block max reduction ----
    s_red[t] = lmax;
    __syncthreads();
    for (int off = 128; off > 0; off >>= 1) {
        if (t < off) s_red[t] = fmaxf(s_red[t], s_red[t + off]);
        __syncthreads();
    }
    const float mx = s_red[0];
    __syncthreads();                     // everyone read mx before s_red reuse

    // ---- exp + block sum reduction (weights left unnormalized in LDS) ----
    float lsum = 0.0f;
    for (int n = t; n < N_TOK; n += 256) {
        float w = __expf(s_w[n] - mx);
        s_w[n] = w;
        lsum += w;
    }
    s_red[t] = lsum;
    __syncthreads();
    for (int off = 128; off > 0; off >>= 1) {
        if (t < off) s_red[t] += s_red[t + off];
        __syncthreads();
    }
    const float inv = 1.0f / s_red[0];
    __syncthreads();

    // ---- Pass C: pooled[i] = inv * sum_n w[n]*x[b,n,i] ----
    // thread t = chunk c (n-range) x component i; weights broadcast from LDS.
    const int i = t & 63;
    const int c = t >> 6;                // 4 chunks of 1024 tokens
    float acc = 0.0f;
    const int nbeg = c * (N_TOK / 4);
    const int nend = nbeg + (N_TOK / 4);
    for (int n = nbeg; n < nend; ++n)
        acc += s_w[n] * xb[n * IN_DIM + i];   // coalesced over i, L2 hit
    s_part[c][i] = acc;
    __syncthreads();
    if (t < IN_DIM) {
        float p = s_part[0][t] + s_part[1][t] + s_part[2][t] + s_part[3][t];
        pooled[(b * NQ + q) * IN_DIM + t] = p * inv;
    }
}

// ---------------------------------------------------------------------------
// Kernel 3: out[m][n] = sum_k pooled[m][k] * W[n][k] + bias[n]
//   M = 128 (b*4+q), N = 768, K = 64.  V_WMMA_F32_16X16X4_F32, one wave per
//   16x16 output tile, 16 unrolled WMMAs over K.
// A frag (16x4 f32): lanes 0-15 hold M=lane, {K,K+1}; lanes 16-31 {K+2,K+3}.
// B frag (4x16 f32): column n striped the same way (B loaded as W row-major,
//   i.e. already K-contiguous per output column).
// D (16x16 f32, 8 VGPRs): vgpr r -> M = r (+8 for lanes 16-31), N = lane&15.
// ---------------------------------------------------------------------------
__global__ void lap_outgemm_kernel(const float* __restrict__ pooled, // [128][64]
                                   const float* __restrict__ W,      // [768][64]
                                   const float* __restrict__ bias,   // [768]
                                   float* __restrict__ out)          // [128][768]
{
    const int n0   = blockIdx.x * 16;
    const int m0   = blockIdx.y * 16;
    const int lane = threadIdx.x;        // 0..31, EXEC all ones
    const int hf   = lane >> 4;          // 0: K,K+1   1: K+2,K+3
    const int l    = lane & 15;

    const int arow = m0 + l;
    const int bcol = n0 + l;
    const float* ap = pooled + arow * IN_DIM;
    const float* bp = W + bcol * IN_DIM;

    v8f acc = {};
    #pragma unroll
    for (int k0 = 0; k0 < IN_DIM; k0 += 4) {
        const int ka = k0 + hf * 2;
        v2f a, bfr;
        a.x   = ap[ka];
        a.y   = ap[ka + 1];
        bfr.x = bp[ka];
        bfr.y = bp[ka + 1];
        acc = __builtin_amdgcn_wmma_f32_16x16x4_f32(
                  /*neg_a=*/false, a, /*neg_b=*/false, bfr,
                  /*c_mod=*/(short)0, acc, /*reuse_a=*/false, /*reuse_b=*/false);
    }

    const int col = n0 + l;
    const float bv = bias[col];
    #pragma unroll
    for (int r = 0; r < 8; ++r) {
        const int row = m0 + hf * 8 + r;
        out[row * FEAT_DIM + col] = acc[r] + bv;
    }
}

// ---------------------------------------------------------------------------
extern "C" void kernel_launch(void* const* d_in, const int* in_sizes, int n_in,
                              void* d_out, int out_size, void* d_ws, size_t ws_size,
                              hipStream_t stream) {
    const float* x    = (const float*)d_in[0];  // dense_features [32,4096,64]
    const int*   mask = (const int*)  d_in[1];  // attention_mask [32,4096]
    const float* qt   = (const float*)d_in[2];  // query_tokens   [4,768]
    const float* W    = (const float*)d_in[3];  // W_proj         [768,64]
    const float* bias = (const float*)d_in[4];  // b_proj         [768]
    float* out = (float*)d_out;                 // [32,4,768]

    const int B = in_sizes[0] / (N_TOK * IN_DIM);   // 32

    float* ws     = (float*)d_ws;
    float* qp     = ws;            // 4*64   = 256 floats
    float* qb     = ws + 256;      // 4 floats
    float* pooled = ws + 512;      // B*4*64 = 8192 floats (256B-aligned)

    lap_qproj_kernel<<<1, 256, 0, stream>>>(qt, W, bias, qp, qb);
    lap_pool_kernel<<<dim3(B, NQ), 256, 0, stream>>>(x, mask, qp, qb, pooled);
    lap_outgemm_kernel<<<dim3(FEAT_DIM / 16, (B * NQ) / 16), 32, 0, stream>>>(
        pooled, W, bias, out);
}